// dot_attention_40742059769887
// MI455X (gfx1250) — compile-verified
//
#include <hip/hip_runtime.h>

typedef __attribute__((ext_vector_type(2))) float v2f;
typedef __attribute__((ext_vector_type(8))) float v8f;

#define TOPK   30
#define BHN    32
#define SEQ    2048
#define HD     64
#define ROWPAD 2052   // padded LDS row stride (floats): kills bank conflicts on tile stores
#define NEG_BIG (-3.402823466e38f)

// wave32 butterfly reductions via ds_swizzle (xor mode, imm offset)
__device__ __forceinline__ float wave_red_max(float x) {
#define STEPM(mk) { float o = __uint_as_float((unsigned)__builtin_amdgcn_ds_swizzle( \
      (int)__float_as_uint(x), ((mk) << 10) | 0x1f)); x = fmaxf(x, o); }
  STEPM(16) STEPM(8) STEPM(4) STEPM(2) STEPM(1)
#undef STEPM
  return x;
}
__device__ __forceinline__ float wave_red_add(float x) {
#define STEPA(mk) { float o = __uint_as_float((unsigned)__builtin_amdgcn_ds_swizzle( \
      (int)__float_as_uint(x), ((mk) << 10) | 0x1f)); x += o; }
  STEPA(16) STEPA(8) STEPA(4) STEPA(2) STEPA(1)
#undef STEPA
  return x;
}

__global__ __launch_bounds__(512)
void topk_attn_f32_wmma_kernel(const float* __restrict__ q,
                               const float* __restrict__ k,
                               const float* __restrict__ v,
                               float* __restrict__ ctx,
                               float* __restrict__ attn)
{
  extern __shared__ float sc[];            // [16][ROWPAD] fp32 scores for this q-tile
  const int tid  = threadIdx.x;
  const int lane = tid & 31;
  const int wave = tid >> 5;               // 0..15
  const int qt   = blockIdx.x;             // 0..127 (query tile)
  const int bh   = blockIdx.y;             // 0..31

  const int r = lane & 15;                 // row-in-tile / N-in-tile
  const int h = lane >> 4;                 // half selector

  // ---------------- Phase 1: scores[16 x 2048] = Qtile * K^T via f32 WMMA ----
  // A (16x4 f32) and B (4x16 f32 = Ktile^T) fragments share the same per-lane
  // indexing into their row-major [16 x 64] source tiles:
  //   frag[j] = tile[lane&15][4*s + 2*(lane>>4) + j],  j = 0,1
  const float* qbase = q + (((size_t)bh * SEQ + (size_t)qt * 16 + r) * HD) + 2 * h;
  v2f aq[16];
#pragma unroll
  for (int s = 0; s < 16; ++s) aq[s] = *(const v2f*)(qbase + 4 * s);

  for (int kt = wave; kt < SEQ / 16; kt += 16) {
    const float* kbase = k + (((size_t)bh * SEQ + (size_t)kt * 16 + r) * HD) + 2 * h;
    v2f bk[16];
#pragma unroll
    for (int s = 0; s < 16; ++s) bk[s] = *(const v2f*)(kbase + 4 * s);

    v8f acc = {0.f, 0.f, 0.f, 0.f, 0.f, 0.f, 0.f, 0.f};
#pragma unroll
    for (int s = 0; s < 16; ++s)
      acc = __builtin_amdgcn_wmma_f32_16x16x4_f32(false, aq[s], false, bk[s],
                                                  (short)0, acc, false, false);
    // D layout: VGPR i, lanes 0-15 -> M=i, N=lane; lanes 16-31 -> M=i+8, N=lane-16
#pragma unroll
    for (int i = 0; i < 8; ++i)
      sc[(size_t)(i + 8 * h) * ROWPAD + kt * 16 + r] = acc[i];
  }
  __syncthreads();

  // ---------------- Phase 2: per-row top-k threshold + masked softmax --------
  const int m  = wave;                     // row within tile, one wave per row
  const int mg = qt * 16 + m;              // global query row

  // load 64 scores/lane, encode as order-preserving u32 keys
  unsigned key[64];
#pragma unroll
  for (int s = 0; s < 64; ++s) {
    unsigned u = __float_as_uint(sc[(size_t)m * ROWPAD + s * 32 + lane]);
    key[s] = u ^ ((unsigned)(((int)u) >> 31) | 0x80000000u);
  }

  // binary search for the key of the TOPK-th largest element:
  // smallest x with count(key > x) < TOPK
  unsigned lo = 0u, hi = 0xFFFFFFFFu;
  for (int it = 0; it < 32 && lo < hi; ++it) {
    unsigned mid = lo + ((hi - lo) >> 1);
    int cnt = 0;
#pragma unroll
    for (int s = 0; s < 64; ++s)
      cnt += __popc(__builtin_amdgcn_ballot_w32(key[s] > mid));
    if (cnt < TOPK) hi = mid; else lo = mid + 1;
  }
  const unsigned kth = lo;

  // masked row max
  float mx = NEG_BIG;
#pragma unroll
  for (int s = 0; s < 64; ++s) {
    unsigned kk = key[s];
    unsigned u  = (kk & 0x80000000u) ? (kk ^ 0x80000000u) : ~kk;  // decode
    float val   = __uint_as_float(u);
    mx = fmaxf(mx, (kk >= kth) ? val : NEG_BIG);
  }
  mx = wave_red_max(mx);

  // exp + sum over kept
  float w[64];
  float sum = 0.f;
#pragma unroll
  for (int s = 0; s < 64; ++s) {
    unsigned kk = key[s];
    unsigned u  = (kk & 0x80000000u) ? (kk ^ 0x80000000u) : ~kk;
    float val   = __uint_as_float(u);
    float e     = (kk >= kth) ? __expf(val - mx) : 0.f;
    w[s] = e;
    sum += e;
  }
  sum = wave_red_add(sum);
  const float inv = 1.0f / sum;

  // stream the dense attn row (537 MB total across grid) with NT stores
  float* arow = attn + ((size_t)bh * SEQ + (size_t)mg) * (size_t)SEQ;
#pragma unroll
  for (int s = 0; s < 64; ++s)
    __builtin_nontemporal_store(w[s] * inv, arow + s * 32 + lane);

  // context = attn_row @ V : only ~TOPK nonzero columns
  float a0 = 0.f, a1 = 0.f;
#pragma unroll
  for (int s = 0; s < 64; ++s) {
    unsigned mset = __builtin_amdgcn_ballot_w32(w[s] > 0.f);
    while (mset) {
      int l = __builtin_ctz(mset); mset &= (mset - 1u);
      float wt = __uint_as_float((unsigned)__builtin_amdgcn_readlane(
                     (int)__float_as_uint(w[s]), l)) * inv;
      int c = s * 32 + l;
      const float* vr = v + ((size_t)bh * SEQ + (size_t)c) * HD;
      a0 += wt * vr[lane];
      a1 += wt * vr[lane + 32];
    }
  }
  float* crow = ctx + ((size_t)bh * SEQ + (size_t)mg) * (size_t)HD;
  crow[lane]      = a0;
  crow[lane + 32] = a1;
}

extern "C" void kernel_launch(void* const* d_in, const int* in_sizes, int n_in,
                              void* d_out, int out_size, void* d_ws, size_t ws_size,
                              hipStream_t stream) {
  (void)in_sizes; (void)n_in; (void)d_ws; (void)ws_size; (void)out_size;
  const float* q = (const float*)d_in[0];
  const float* k = (const float*)d_in[1];
  const float* v = (const float*)d_in[2];
  // d_in[3]=B, d_in[4]=num_heads : unused by the math

  float* ctx  = (float*)d_out;                                   // [32,2048,64]
  float* attn = ctx + (size_t)BHN * SEQ * HD;                    // [32,2048,2048]

  const int lds_bytes = 16 * ROWPAD * 4;                         // 131,328 B
  hipFuncSetAttribute((const void*)topk_attn_f32_wmma_kernel,
                      hipFuncAttributeMaxDynamicSharedMemorySize, lds_bytes);

  dim3 grid(SEQ / 16, BHN);   // 128 x 32 blocks
  topk_attn_f32_wmma_kernel<<<grid, 512, lds_bytes, stream>>>(q, k, v, ctx, attn);
}